// GCN_15616501088483
// MI455X (gfx1250) — compile-verified
//
#include <hip/hip_runtime.h>
#include <hip/hip_bf16.h>

typedef float v2f __attribute__((ext_vector_type(2)));
typedef float v8f __attribute__((ext_vector_type(8)));

#define WAVES 8           // 256-thread blocks = 8 wave32s
#define HID 10
#define EDGE_DIM 6

__device__ __forceinline__ float lrelu(float x) {
    // leaky_relu(x, 0.1) == max(x, 0.1*x)
    return fmaxf(x, 0.1f * x);
}

// Stage a [rows x 10] weight matrix into LDS as [rows_pad x 16], zero-padded
// in both K (rows >= rows_real) and N (cols 10..15).  Unconditional B loads.
__device__ __forceinline__ void stage_weights_padded(float* wlds, const float* w,
                                                     int rows_real, int rows_pad,
                                                     int t, int nthreads) {
    for (int i = t; i < rows_pad * 16; i += nthreads) {
        int row = i >> 4, col = i & 15;
        wlds[i] = (col < HID && row < rows_real) ? w[row * HID + col] : 0.0f;
    }
}

// -------------------------------------------------------------------------
// h = lrelu(vertex[N,82] @ w_init[82,10])   (K padded 82 -> 84)
// -------------------------------------------------------------------------
__global__ void gnn_embed_kernel(const float* __restrict__ vertex,
                                 const float* __restrict__ w_init,
                                 float* __restrict__ h_out, int n_nodes) {
    __shared__ float wlds[84 * 16];           // [84][16] zero-padded
    __shared__ float alds[WAVES][16 * 85];    // 16 nodes x 84 feats, stride 85
    int t = threadIdx.x;
    stage_weights_padded(wlds, w_init, 82, 84, t, blockDim.x);
    __syncthreads();

    int wave = t >> 5, lane = t & 31;
    int tile = blockIdx.x * WAVES + wave;
    int base = tile * 16;
    if (base >= n_nodes) return;              // wave-uniform exit

    float* A = alds[wave];
    {   // stage: 2 lanes per node row, 41 contiguous floats each
        int row = lane & 15, half = lane >> 4;
        int node = base + row;
        const float* vs = vertex + (size_t)node * 82;
        for (int j = 0; j < 41; ++j) {
            int k = half * 41 + j;
            A[row * 85 + k] = (node < n_nodes) ? vs[k] : 0.0f;
        }
        if (half == 0) { A[row * 85 + 82] = 0.0f; A[row * 85 + 83] = 0.0f; }
    }
    __builtin_amdgcn_wave_barrier();

    v8f c = {};
    int m = lane & 15;                        // A: row M ; B/D: col N
    int koff = (lane < 16) ? 0 : 2;
    for (int kc = 0; kc < 21; ++kc) {         // K = 84 = 21 * 4
        int kb = kc * 4 + koff;
        v2f a, b;
        a.x = A[m * 85 + kb];
        a.y = A[m * 85 + kb + 1];
        b.x = wlds[kb * 16 + m];              // unconditional (zero-padded)
        b.y = wlds[(kb + 1) * 16 + m];
        c = __builtin_amdgcn_wmma_f32_16x16x4_f32(false, a, false, b,
                                                  (short)0, c, false, false);
    }
    if (m < HID) {
        int rbase = (lane < 16) ? 0 : 8;
#pragma unroll
        for (int r = 0; r < 8; ++r) {
            int node = base + rbase + r;
            if (node < n_nodes)
                h_out[(size_t)node * HID + m] = lrelu(c[r]);
        }
    }
}

// -------------------------------------------------------------------------
// zero fill
// -------------------------------------------------------------------------
__global__ void gnn_zero_kernel(float* __restrict__ p, int n) {
    int i = blockIdx.x * blockDim.x + threadIdx.x;
    if (i < n) p[i] = 0.0f;
}

// -------------------------------------------------------------------------
// msg = lrelu([h[src] | edge_feat] @ g_w[16,10]);  agg[dst] += msg
// -------------------------------------------------------------------------
__global__ void gnn_edge_kernel(const float* __restrict__ h_in,
                                const float* __restrict__ efeat,
                                const int* __restrict__ src,
                                const int* __restrict__ dst,
                                const float* __restrict__ g_w,
                                float* __restrict__ agg, int n_edges) {
    __shared__ float wlds[16 * 16];           // [16][16] zero-padded
    __shared__ float alds[WAVES][16 * 17];    // 16 edges x 16 feats, stride 17
    __shared__ int   slds[WAVES][16];
    __shared__ int   dlds[WAVES][16];
    int t = threadIdx.x;
    stage_weights_padded(wlds, g_w, 16, 16, t, blockDim.x);
    __syncthreads();

    int wave = t >> 5, lane = t & 31;
    long long tile = (long long)blockIdx.x * WAVES + wave;
    long long base = tile * 16;
    if (base >= n_edges) return;              // wave-uniform exit

    float* A = alds[wave];
    int* S = slds[wave];
    int* D = dlds[wave];
    if (lane < 16) {
        long long e = base + lane;
        bool ok = e < (long long)n_edges;
        S[lane] = ok ? src[e] : 0;
        D[lane] = ok ? dst[e] : 0;
    }
    __builtin_amdgcn_wave_barrier();

    // stage A: feat k<10 from gathered h[src], k in [10,16) from edge_feat
    for (int i = lane; i < 256; i += 32) {
        int e = i >> 4, k = i & 15;
        long long E = base + e;
        float v = 0.0f;
        if (E < (long long)n_edges) {
            if (k < HID) v = h_in[(size_t)S[e] * HID + k];
            else         v = efeat[(size_t)E * EDGE_DIM + (k - HID)];
        }
        A[e * 17 + k] = v;
    }
    __builtin_amdgcn_wave_barrier();

    v8f c = {};
    int m = lane & 15;
    int koff = (lane < 16) ? 0 : 2;
#pragma unroll
    for (int kc = 0; kc < 4; ++kc) {          // K = 16
        int kb = kc * 4 + koff;
        v2f a, b;
        a.x = A[m * 17 + kb];
        a.y = A[m * 17 + kb + 1];
        b.x = wlds[kb * 16 + m];
        b.y = wlds[(kb + 1) * 16 + m];
        c = __builtin_amdgcn_wmma_f32_16x16x4_f32(false, a, false, b,
                                                  (short)0, c, false, false);
    }
    if (m < HID) {
        int rbase = (lane < 16) ? 0 : 8;
#pragma unroll
        for (int r = 0; r < 8; ++r) {
            long long E = base + rbase + r;
            if (E < (long long)n_edges)
                atomicAdd(&agg[(size_t)D[rbase + r] * HID + m], lrelu(c[r]));
        }
    }
}

// -------------------------------------------------------------------------
// h_new = lrelu([h | agg] @ u_w[20,10])
// -------------------------------------------------------------------------
__global__ void gnn_update_kernel(const float* __restrict__ h_in,
                                  const float* __restrict__ agg,
                                  const float* __restrict__ u_w,
                                  float* __restrict__ h_out, int n_nodes) {
    __shared__ float wlds[20 * 16];           // [20][16] zero-padded
    __shared__ float alds[WAVES][16 * 21];    // 16 nodes x 20 feats, stride 21
    int t = threadIdx.x;
    stage_weights_padded(wlds, u_w, 20, 20, t, blockDim.x);
    __syncthreads();

    int wave = t >> 5, lane = t & 31;
    int tile = blockIdx.x * WAVES + wave;
    int base = tile * 16;
    if (base >= n_nodes) return;              // wave-uniform exit

    float* A = alds[wave];
    {   // half 0: h row (k 0..9), half 1: agg row (k 10..19)
        int row = lane & 15, half = lane >> 4;
        int node = base + row;
        const float* p = (half == 0) ? (h_in + (size_t)node * HID)
                                     : (agg  + (size_t)node * HID);
        int k0 = half * HID;
#pragma unroll
        for (int j = 0; j < HID; ++j)
            A[row * 21 + k0 + j] = (node < n_nodes) ? p[j] : 0.0f;
    }
    __builtin_amdgcn_wave_barrier();

    v8f c = {};
    int m = lane & 15;
    int koff = (lane < 16) ? 0 : 2;
#pragma unroll
    for (int kc = 0; kc < 5; ++kc) {          // K = 20
        int kb = kc * 4 + koff;
        v2f a, b;
        a.x = A[m * 21 + kb];
        a.y = A[m * 21 + kb + 1];
        b.x = wlds[kb * 16 + m];
        b.y = wlds[(kb + 1) * 16 + m];
        c = __builtin_amdgcn_wmma_f32_16x16x4_f32(false, a, false, b,
                                                  (short)0, c, false, false);
    }
    if (m < HID) {
        int rbase = (lane < 16) ? 0 : 8;
#pragma unroll
        for (int r = 0; r < 8; ++r) {
            int node = base + rbase + r;
            if (node < n_nodes)
                h_out[(size_t)node * HID + m] = lrelu(c[r]);
        }
    }
}

// -------------------------------------------------------------------------
extern "C" void kernel_launch(void* const* d_in, const int* in_sizes, int n_in,
                              void* d_out, int out_size, void* d_ws, size_t ws_size,
                              hipStream_t stream) {
    const float* vertex = (const float*)d_in[0];
    const float* efeat  = (const float*)d_in[1];
    const int*   src    = (const int*)d_in[2];
    const int*   dst    = (const int*)d_in[3];
    const float* w_init = (const float*)d_in[4];
    const float* g_w    = (const float*)d_in[5];
    const float* u_w    = (const float*)d_in[6];
    float* out = (float*)d_out;

    const int N = in_sizes[0] / 82;           // 200000
    const int E = in_sizes[2];                // 6400000
    const int NH = N * HID;

    float* hA  = (float*)d_ws;                // [N,10]
    float* hB  = hA + (size_t)NH;             // [N,10]
    float* agg = hB + (size_t)NH;             // [N,10]

    dim3 blk(256);
    int node_tiles  = (N + 15) / 16;
    int node_blocks = (node_tiles + WAVES - 1) / WAVES;
    int edge_tiles  = (E + 15) / 16;
    int edge_blocks = (edge_tiles + WAVES - 1) / WAVES;
    int zero_blocks = (NH + 255) / 256;

    gnn_embed_kernel<<<node_blocks, blk, 0, stream>>>(vertex, w_init, hA, N);

    // layer 0
    gnn_zero_kernel<<<zero_blocks, blk, 0, stream>>>(agg, NH);
    gnn_edge_kernel<<<edge_blocks, blk, 0, stream>>>(hA, efeat, src, dst, g_w, agg, E);
    gnn_update_kernel<<<node_blocks, blk, 0, stream>>>(hA, agg, u_w, hB, N);

    // layer 1
    gnn_zero_kernel<<<zero_blocks, blk, 0, stream>>>(agg, NH);
    gnn_edge_kernel<<<edge_blocks, blk, 0, stream>>>(hB, efeat, src, dst, g_w, agg, E);
    gnn_update_kernel<<<node_blocks, blk, 0, stream>>>(hB, agg, u_w, out, N);
}